// GC_53695681134706
// MI455X (gfx1250) — compile-verified
//
#include <hip/hip_runtime.h>

typedef __attribute__((ext_vector_type(2))) float v2f;
typedef __attribute__((ext_vector_type(8))) float v8f;

#define D_FEAT 64

// ---------------- zero workspace ----------------
__global__ void GC_zero_kernel(float* __restrict__ p, long long n) {
    long long i = (long long)blockIdx.x * blockDim.x + threadIdx.x;
    if (i < n) p[i] = 0.0f;
}

// ---------------- degree counts ----------------
__global__ void GC_degree_kernel(const int* __restrict__ src, const int* __restrict__ dst,
                                 int* __restrict__ deg_out, int* __restrict__ deg_in, int e) {
    int i = blockIdx.x * blockDim.x + threadIdx.x;
    if (i < e) {
        atomicAdd(&deg_out[src[i]], 1);
        atomicAdd(&deg_in[dst[i]], 1);
    }
}

// ---------------- rsqrt scales ----------------
__global__ void GC_scale_kernel(const int* __restrict__ deg_out, const int* __restrict__ deg_in,
                                float* __restrict__ s_src, float* __restrict__ s_dst, int n) {
    int i = blockIdx.x * blockDim.x + threadIdx.x;
    if (i < n) {
        s_src[i] = rsqrtf(fmaxf((float)deg_out[i], 1.0f));
        s_dst[i] = rsqrtf(fmaxf((float)deg_in[i],  1.0f));
    }
}

// ---------------- edge gather + scatter-add ----------------
// 16 threads per edge, each handles 4 contiguous floats (float4 gather, 4 atomic f32 adds).
__global__ void GC_scatter_kernel(const float* __restrict__ feat, const int* __restrict__ src,
                                  const int* __restrict__ dst, const float* __restrict__ s_src,
                                  float* __restrict__ agg, int e) {
    long long gid = (long long)blockIdx.x * blockDim.x + threadIdx.x;
    int edge = (int)(gid >> 4);
    int c    = ((int)gid & 15) * 4;
    if (edge < e) {
        int s = src[edge];
        int d = dst[edge];
        float sc = s_src[s];
        const float4* fp = (const float4*)(feat + (long long)s * D_FEAT + c);
        float4 v = *fp;
        float* o = agg + (long long)d * D_FEAT + c;
        atomicAdd(o + 0, v.x * sc);
        atomicAdd(o + 1, v.y * sc);
        atomicAdd(o + 2, v.z * sc);
        atomicAdd(o + 3, v.w * sc);
    }
}

// ---------------- GEMM: out = (agg @ W^T + b) * s_dst, via V_WMMA_F32_16X16X4_F32 ----------------
// Block = 256 threads = 8 waves; each wave computes a 16(node) x 64(feature) tile.
__global__ void GC_gemm_kernel(const float* __restrict__ agg, const float* __restrict__ W,
                               const float* __restrict__ b, const float* __restrict__ s_dst,
                               float* __restrict__ out, int n) {
    __shared__ float Wl[D_FEAT * D_FEAT];   // 16 KB: W row-major, W[nout*64 + k]
    for (int i = threadIdx.x; i < D_FEAT * D_FEAT; i += blockDim.x) Wl[i] = W[i];
    __syncthreads();

    int wave = threadIdx.x >> 5;     // 0..7
    int lane = threadIdx.x & 31;
    int half = lane >> 4;            // 0/1 -> which K-pair (A) / K-pair (B) / M-half (C)
    int l16  = lane & 15;

    long long base = ((long long)blockIdx.x * 8 + wave) * 16;
    if (base >= n) return;           // whole wave exits together (EXEC stays all-1 for WMMA)

    v8f acc0 = {}, acc1 = {}, acc2 = {}, acc3 = {};

    // A-fragment row for this lane (M = l16); clamp row for the (non-existent) ragged tail
    long long arow = base + l16;
    if (arow >= n) arow = n - 1;
    const float* aptr = agg + arow * D_FEAT;

    #pragma unroll
    for (int kk = 0; kk < D_FEAT; kk += 4) {
        v2f a;                                   // A[16x4]: lane half selects K pair
        a.x = aptr[kk + 2 * half + 0];
        a.y = aptr[kk + 2 * half + 1];

        v2f bf0, bf1, bf2, bf3;                  // B[4x16] per N-tile: B[k][nn] = W[nn][k]
        {
            int k0 = kk + 2 * half;
            bf0.x = Wl[( 0 + l16) * D_FEAT + k0]; bf0.y = Wl[( 0 + l16) * D_FEAT + k0 + 1];
            bf1.x = Wl[(16 + l16) * D_FEAT + k0]; bf1.y = Wl[(16 + l16) * D_FEAT + k0 + 1];
            bf2.x = Wl[(32 + l16) * D_FEAT + k0]; bf2.y = Wl[(32 + l16) * D_FEAT + k0 + 1];
            bf3.x = Wl[(48 + l16) * D_FEAT + k0]; bf3.y = Wl[(48 + l16) * D_FEAT + k0 + 1];
        }
        acc0 = __builtin_amdgcn_wmma_f32_16x16x4_f32(false, a, false, bf0, (short)0, acc0, false, false);
        acc1 = __builtin_amdgcn_wmma_f32_16x16x4_f32(false, a, false, bf1, (short)0, acc1, false, false);
        acc2 = __builtin_amdgcn_wmma_f32_16x16x4_f32(false, a, false, bf2, (short)0, acc2, false, false);
        acc3 = __builtin_amdgcn_wmma_f32_16x16x4_f32(false, a, false, bf3, (short)0, acc3, false, false);
    }

    // D layout: N = l16 (+16*t), M = v + 8*half. Fused bias + in-degree scale.
    v8f accs[4] = {acc0, acc1, acc2, acc3};
    #pragma unroll
    for (int t = 0; t < 4; ++t) {
        int ng = t * 16 + l16;
        float bias = b[ng];
        #pragma unroll
        for (int v = 0; v < 8; ++v) {
            long long m = base + v + 8 * half;
            if (m < n) out[m * D_FEAT + ng] = (accs[t][v] + bias) * s_dst[m];
        }
    }
}

extern "C" void kernel_launch(void* const* d_in, const int* in_sizes, int n_in,
                              void* d_out, int out_size, void* d_ws, size_t ws_size,
                              hipStream_t stream) {
    const float* feat = (const float*)d_in[0];
    const int*   src  = (const int*)  d_in[1];
    const int*   dst  = (const int*)  d_in[2];
    const float* W    = (const float*)d_in[3];
    const float* b    = (const float*)d_in[4];
    float* out = (float*)d_out;

    const int nE = in_sizes[1];
    const int nN = in_sizes[0] / D_FEAT;

    // workspace layout: [agg: nN*64 f32][deg_out: nN i32][deg_in: nN i32][s_src: nN f32][s_dst: nN f32]
    float* agg     = (float*)d_ws;
    int*   deg_out = (int*)(agg + (size_t)nN * D_FEAT);
    int*   deg_in  = deg_out + nN;
    float* s_src   = (float*)(deg_in + nN);
    float* s_dst   = s_src + nN;

    // zero agg + both degree arrays (contiguous prefix of workspace) every call
    long long zn = (long long)nN * D_FEAT + 2LL * nN;
    GC_zero_kernel<<<(unsigned)((zn + 255) / 256), 256, 0, stream>>>((float*)d_ws, zn);

    GC_degree_kernel<<<(nE + 255) / 256, 256, 0, stream>>>(src, dst, deg_out, deg_in, nE);
    GC_scale_kernel<<<(nN + 255) / 256, 256, 0, stream>>>(deg_out, deg_in, s_src, s_dst, nN);

    long long st = (long long)nE * 16;   // 16 threads per edge
    GC_scatter_kernel<<<(unsigned)((st + 255) / 256), 256, 0, stream>>>(feat, src, dst, s_src, agg, nE);

    GC_gemm_kernel<<<(nN + 127) / 128, 256, 0, stream>>>(agg, W, b, s_dst, out, nN);
}